// SolarOptimizationLoss_77592879169759
// MI455X (gfx1250) — compile-verified
//
#include <hip/hip_runtime.h>
#include <hip/hip_bf16.h>
#include <math.h>

#define N_ELEMS 8192
#define TILE    256
#define NTILE   (N_ELEMS / TILE)            // 32
#define NBLK    (NTILE * (NTILE + 1) / 2)   // 528 upper-triangular tile pairs

typedef __attribute__((ext_vector_type(2))) float v2f;
typedef __attribute__((ext_vector_type(8))) float v8f;

// Half-wave sum on the CDNA5 matrix unit (exact fp32, layout-agnostic B=ones).
// A (16x4 f32): lanes 0-15 -> {VGPR0=K0, VGPR1=K1}, lanes 16-31 -> {K2, K3}.
// With A.x = p, A.y = 0, B = ones: D[m][n] = p[m] + p[m+16].
// Summing a lane's 8 D registers gives: lanes 0-15 -> sum over m=0..7,
// lanes 16-31 -> sum over m=8..15. Lane0 + Lane16 = full 32-lane total.
__device__ __forceinline__ float wave_halfsum_wmma(float p) {
  v2f a; a.x = p;    a.y = 0.0f;
  v2f b; b.x = 1.0f; b.y = 1.0f;
  v8f c = {};
  v8f d = __builtin_amdgcn_wmma_f32_16x16x4_f32(
      /*neg_a=*/false, a, /*neg_b=*/false, b,
      /*c_mod=*/(short)0, c, /*reuse_a=*/false, /*reuse_b=*/false);
  return d[0] + d[1] + d[2] + d[3] + d[4] + d[5] + d[6] + d[7];
}

__global__ void __launch_bounds__(256)
rank_tiles_kernel(const float* __restrict__ score, const int* __restrict__ rank,
                  float* __restrict__ bsum, float* __restrict__ bcnt) {
  __shared__ __align__(16) float s_col[TILE];
  __shared__ __align__(16) int   r_col[TILE];
  __shared__ float s_row[TILE];
  __shared__ int   r_row[TILE];
  __shared__ float wsum_lo[8], wsum_hi[8], wcnt_lo[8], wcnt_hi[8];

  const int tid = threadIdx.x;

  // Map blockIdx.x -> (bi, bj), bi <= bj, over the 32x32 tile upper triangle.
  int k = blockIdx.x;
  int bi = 0;
  while (k >= (NTILE - bi)) { k -= (NTILE - bi); ++bi; }
  const int bj = bi + k;

  s_row[tid] = score[bi * TILE + tid]; r_row[tid] = rank[bi * TILE + tid];
  s_col[tid] = score[bj * TILE + tid]; r_col[tid] = rank[bj * TILE + tid];
  __syncthreads();

  const float ci  = 1.0f - s_row[tid];   // hinge = max(ci + s_j, 0)
  const int   r_i = r_row[tid];
  float sum = 0.0f;
  float cnt = 0.0f;                       // exact: per-thread <= 256 < 2^24

  if (bi == bj) {                         // diagonal tile: need j > tid
#pragma unroll 2
    for (int j = 0; j < TILE; j += 4) {
      const float4 s4 = *(const float4*)&s_col[j];
      const int4   r4 = *(const int4*)&r_col[j];
      if ((j + 0 > tid) && (r_i < r4.x)) { sum += fmaxf(ci + s4.x, 0.0f); cnt += 1.0f; }
      if ((j + 1 > tid) && (r_i < r4.y)) { sum += fmaxf(ci + s4.y, 0.0f); cnt += 1.0f; }
      if ((j + 2 > tid) && (r_i < r4.z)) { sum += fmaxf(ci + s4.z, 0.0f); cnt += 1.0f; }
      if ((j + 3 > tid) && (r_i < r4.w)) { sum += fmaxf(ci + s4.w, 0.0f); cnt += 1.0f; }
    }
  } else {                                // off-diagonal: all columns have j > i
#pragma unroll 2
    for (int j = 0; j < TILE; j += 4) {
      const float4 s4 = *(const float4*)&s_col[j];
      const int4   r4 = *(const int4*)&r_col[j];
      if (r_i < r4.x) { sum += fmaxf(ci + s4.x, 0.0f); cnt += 1.0f; }
      if (r_i < r4.y) { sum += fmaxf(ci + s4.y, 0.0f); cnt += 1.0f; }
      if (r_i < r4.z) { sum += fmaxf(ci + s4.z, 0.0f); cnt += 1.0f; }
      if (r_i < r4.w) { sum += fmaxf(ci + s4.w, 0.0f); cnt += 1.0f; }
    }
  }

  // Wave-level reductions on the matrix unit (EXEC all-1s here; wave32).
  const float hs = wave_halfsum_wmma(sum);
  const float hc = wave_halfsum_wmma(cnt);

  const int lane = tid & 31;
  const int wid  = tid >> 5;
  if (lane == 0)  { wsum_lo[wid] = hs; wcnt_lo[wid] = hc; }
  if (lane == 16) { wsum_hi[wid] = hs; wcnt_hi[wid] = hc; }
  __syncthreads();

  if (tid == 0) {
    float bs = 0.0f, bc = 0.0f;
    for (int w = 0; w < 8; ++w) {
      bs += wsum_lo[w] + wsum_hi[w];
      bc += wcnt_lo[w] + wcnt_hi[w];    // integer-valued, <= 65536: exact
    }
    bsum[blockIdx.x] = bs;
    bcnt[blockIdx.x] = bc;
  }
}

__global__ void __launch_bounds__(256)
finalize_kernel(const float* __restrict__ bsum, const float* __restrict__ bcnt,
                const float* __restrict__ cap_p, const float* __restrict__ cap_g,
                const float* __restrict__ roi_p, const float* __restrict__ roi_g,
                const float* __restrict__ via_p, const float* __restrict__ via_g,
                float* __restrict__ out) {
  __shared__ double sh_rank[256];
  __shared__ double sh_cnt[256];
  __shared__ double sh_cap[256];
  __shared__ double sh_roi[256];
  __shared__ double sh_bce[256];

  const int tid = threadIdx.x;

  double rs = 0.0, rc = 0.0;
  for (int i = tid; i < NBLK; i += 256) {
    rs += (double)bsum[i];
    rc += (double)bcnt[i];              // exact integer accumulation in f64
  }

  const float PLO = 1e-7f;
  const float PHI = (float)(1.0 - 1e-7);
  double caps = 0.0, rois = 0.0, bces = 0.0;
  for (int i = tid; i < N_ELEMS; i += 256) {
    const float dc = cap_p[i] - cap_g[i];
    caps += (double)(dc * dc);

    const float dr = roi_p[i] - roi_g[i];
    const float ad = fabsf(dr);
    const float rl = (ad < 1.0f) ? (0.5f * dr * dr) : (ad - 0.5f);
    rois += (double)rl;

    float p = via_p[i];
    p = fminf(fmaxf(p, PLO), PHI);
    const float g = via_g[i];
    bces += (double)(g * logf(p) + (1.0f - g) * log1pf(-p));
  }

  sh_rank[tid] = rs; sh_cnt[tid] = rc;
  sh_cap[tid] = caps; sh_roi[tid] = rois; sh_bce[tid] = bces;
  __syncthreads();

  for (int off = 128; off > 0; off >>= 1) {
    if (tid < off) {
      sh_rank[tid] += sh_rank[tid + off];
      sh_cnt[tid]  += sh_cnt[tid + off];
      sh_cap[tid]  += sh_cap[tid + off];
      sh_roi[tid]  += sh_roi[tid + off];
      sh_bce[tid]  += sh_bce[tid + off];
    }
    __syncthreads();
  }

  if (tid == 0) {
    const double cnt = sh_cnt[0];
    const float ranking = (cnt > 0.0) ? (float)(sh_rank[0] / cnt) : 0.0f;
    const float cap = (float)(sh_cap[0] / (double)N_ELEMS);
    const float roi = (float)(sh_roi[0] / (double)N_ELEMS);
    const float via = (float)(-sh_bce[0] / (double)N_ELEMS);
    out[0] = 0.25f * (ranking + cap + roi + via);
  }
}

extern "C" void kernel_launch(void* const* d_in, const int* in_sizes, int n_in,
                              void* d_out, int out_size, void* d_ws, size_t ws_size,
                              hipStream_t stream) {
  (void)in_sizes; (void)n_in; (void)out_size; (void)ws_size;

  const float* score = (const float*)d_in[0];
  const int*   rank  = (const int*)d_in[1];
  const float* cap_p = (const float*)d_in[2];
  const float* cap_g = (const float*)d_in[3];
  const float* roi_p = (const float*)d_in[4];
  const float* roi_g = (const float*)d_in[5];
  const float* via_p = (const float*)d_in[6];
  const float* via_g = (const float*)d_in[7];

  float* bsum = (float*)d_ws;
  float* bcnt = (float*)((char*)d_ws + NBLK * sizeof(float));

  rank_tiles_kernel<<<NBLK, 256, 0, stream>>>(score, rank, bsum, bcnt);
  finalize_kernel<<<1, 256, 0, stream>>>(bsum, bcnt, cap_p, cap_g,
                                         roi_p, roi_g, via_p, via_g,
                                         (float*)d_out);
}